// GraphEncoder_26568667693115
// MI455X (gfx1250) — compile-verified
//
#include <hip/hip_runtime.h>
#include <hip/hip_bf16.h>

typedef __attribute__((ext_vector_type(2))) float v2f;
typedef __attribute__((ext_vector_type(8))) float v8f;

#define N_NODES 50000
#define N_EDGES 1600000
#define IN_CH   256
#define HID     128
#define LAT     64

// ---------------- degree / normalization ----------------

__global__ void deg_init_kernel(float* deg, int n) {
    int i = blockIdx.x * blockDim.x + threadIdx.x;
    if (i < n) deg[i] = 1.0f;   // self-loop contributes 1 to every node's degree
}

__global__ void deg_count_kernel(const int* __restrict__ dst, float* deg, int E) {
    int e = blockIdx.x * blockDim.x + threadIdx.x;
    if (e < E) atomicAdd(&deg[dst[e]], 1.0f);
}

__global__ void dinv_kernel(float* deg, int n) {
    int i = blockIdx.x * blockDim.x + threadIdx.x;
    if (i < n) deg[i] = rsqrtf(deg[i]);   // deg >= 1 always (self-loops)
}

// ---------------- fp32 WMMA GEMM: C[M,N] = A[M,K] @ B[K,N] ----------------
// One wave computes a 16-row x N-column stripe using V_WMMA_F32_16X16X4_F32.
// 16x16x4 f32 fragment layout (ISA 7.12.2):
//   A: lane<16 -> M=lane,    holds K=k0+0 (a.x), k0+1 (a.y)
//      lane>=16 -> M=lane-16, holds K=k0+2 (a.x), k0+3 (a.y)
//   B: mirrored with N in place of M
//   C/D: VGPR v -> M = v + 8*(lane>=16), N = lane&15

template<int N, int K>
__global__ void wmma_gemm_kernel(const float* __restrict__ A,
                                 const float* __restrict__ B,
                                 float* __restrict__ C,
                                 int M) {
    constexpr int NT = N / 16;
    int wave = (int)((blockIdx.x * blockDim.x + threadIdx.x) >> 5);
    int lane = threadIdx.x & 31;
    int m0 = wave * 16;
    if (m0 >= M) return;

    int hl = lane >> 4;    // 0 or 1 (which K half)
    int r  = lane & 15;    // row (A) / col (B)

    v8f acc[NT];
#pragma unroll
    for (int t = 0; t < NT; ++t) acc[t] = v8f{};

    const float* arow = A + (size_t)(m0 + r) * K;
    for (int k0 = 0; k0 < K; k0 += 4) {
        v2f a;
        a.x = arow[k0 + 2 * hl + 0];
        a.y = arow[k0 + 2 * hl + 1];
        const float* b0 = B + (size_t)(k0 + 2 * hl) * N;
#pragma unroll
        for (int t = 0; t < NT; ++t) {
            v2f b;
            b.x = b0[t * 16 + r];
            b.y = b0[N + t * 16 + r];
            acc[t] = __builtin_amdgcn_wmma_f32_16x16x4_f32(
                /*neg_a=*/false, a, /*neg_b=*/false, b,
                /*c_mod=*/(short)0, acc[t], /*reuse_a=*/false, /*reuse_b=*/false);
        }
    }

#pragma unroll
    for (int t = 0; t < NT; ++t) {
        float* crow = C + (size_t)m0 * N + t * 16 + r;
#pragma unroll
        for (int v = 0; v < 8; ++v) {
            crow[(size_t)(v + 8 * hl) * N] = acc[t][v];
        }
    }
}

// ---------------- self-loop init: agg[i,:] = dinv[i]^2 * h[i,:] ----------------

template<int C>
__global__ void selfloop_init_kernel(const float* __restrict__ h,
                                     const float* __restrict__ dinv,
                                     float* __restrict__ agg, int n) {
    long long idx = (long long)blockIdx.x * blockDim.x + threadIdx.x;
    long long total = (long long)n * C;
    if (idx >= total) return;
    int i = (int)(idx >> (C == 128 ? 7 : 6));
    float d = dinv[i];
    agg[idx] = d * d * h[idx];
}

// ---------------- edge scatter: agg[dst,:] += dinv[src]*dinv[dst] * h[src,:] ----------------
// Each thread handles 4 channels of one edge (float4 gather + 4 float atomics).

template<int C>
__global__ void edge_scatter_kernel(const float* __restrict__ h,
                                    const float* __restrict__ dinv,
                                    const int* __restrict__ src,
                                    const int* __restrict__ dst,
                                    float* __restrict__ agg, int E) {
    constexpr int TPE = C / 4;   // threads per edge
    long long gid = (long long)blockIdx.x * blockDim.x + threadIdx.x;
    long long e = gid / TPE;
    int c4 = (int)(gid % TPE) * 4;
    if (e >= E) return;
    int s = src[e], d = dst[e];
    float norm = dinv[s] * dinv[d];
    const float4 hv = *(const float4*)(h + (size_t)s * C + c4);
    float* out = agg + (size_t)d * C + c4;
    atomicAdd(out + 0, norm * hv.x);
    atomicAdd(out + 1, norm * hv.y);
    atomicAdd(out + 2, norm * hv.z);
    atomicAdd(out + 3, norm * hv.w);
}

// ---------------- bias + ReLU ----------------

template<int C>
__global__ void bias_relu_kernel(const float* __restrict__ agg,
                                 const float* __restrict__ b,
                                 float* __restrict__ out, int n) {
    long long idx = (long long)blockIdx.x * blockDim.x + threadIdx.x;
    long long total = (long long)n * C;
    if (idx >= total) return;
    int c = (int)(idx & (C - 1));
    out[idx] = fmaxf(agg[idx] + b[c], 0.0f);
}

// ---------------- launcher ----------------

extern "C" void kernel_launch(void* const* d_in, const int* in_sizes, int n_in,
                              void* d_out, int out_size, void* d_ws, size_t ws_size,
                              hipStream_t stream) {
    const float* x  = (const float*)d_in[0];
    const int*   ei = (const int*)d_in[1];       // [2, E]: row0 = src, row1 = dst
    const float* W1 = (const float*)d_in[2];
    const float* b1 = (const float*)d_in[3];
    const float* W2 = (const float*)d_in[4];
    const float* b2 = (const float*)d_in[5];
    float* out = (float*)d_out;

    const int* src = ei;
    const int* dst = ei + N_EDGES;

    // workspace layout (floats); all offsets are multiples of 4 floats (16B aligned)
    float* ws   = (float*)d_ws;
    float* dinv = ws;                                  // N_NODES   (deg -> dinv in place)
    float* h1   = dinv + N_NODES;                      // N_NODES * HID
    float* agg1 = h1 + (size_t)N_NODES * HID;          // N_NODES * HID
    float* h2   = agg1 + (size_t)N_NODES * HID;        // N_NODES * LAT
    float* agg2 = h2 + (size_t)N_NODES * LAT;          // N_NODES * LAT

    const int T = 256;

    // 1) degrees + dinv
    deg_init_kernel<<<(N_NODES + T - 1) / T, T, 0, stream>>>(dinv, N_NODES);
    deg_count_kernel<<<(N_EDGES + T - 1) / T, T, 0, stream>>>(dst, dinv, N_EDGES);
    dinv_kernel<<<(N_NODES + T - 1) / T, T, 0, stream>>>(dinv, N_NODES);

    // 2) layer 1: h1 = x @ W1   (WMMA fp32), aggregate, bias+relu (relu'd result -> h1)
    {
        int waves = N_NODES / 16;                       // 3125
        int blocks = (waves * 32 + T - 1) / T;
        wmma_gemm_kernel<HID, IN_CH><<<blocks, T, 0, stream>>>(x, W1, h1, N_NODES);

        long long tot = (long long)N_NODES * HID;
        selfloop_init_kernel<HID><<<(int)((tot + T - 1) / T), T, 0, stream>>>(h1, dinv, agg1, N_NODES);

        long long sthreads = (long long)N_EDGES * (HID / 4);
        edge_scatter_kernel<HID><<<(int)((sthreads + T - 1) / T), T, 0, stream>>>(
            h1, dinv, src, dst, agg1, N_EDGES);

        bias_relu_kernel<HID><<<(int)((tot + T - 1) / T), T, 0, stream>>>(agg1, b1, h1, N_NODES);
    }

    // 3) layer 2: h2 = h1 @ W2 (WMMA fp32), aggregate, bias+relu -> d_out
    {
        int waves = N_NODES / 16;
        int blocks = (waves * 32 + T - 1) / T;
        wmma_gemm_kernel<LAT, HID><<<blocks, T, 0, stream>>>(h1, W2, h2, N_NODES);

        long long tot = (long long)N_NODES * LAT;
        selfloop_init_kernel<LAT><<<(int)((tot + T - 1) / T), T, 0, stream>>>(h2, dinv, agg2, N_NODES);

        long long sthreads = (long long)N_EDGES * (LAT / 4);
        edge_scatter_kernel<LAT><<<(int)((sthreads + T - 1) / T), T, 0, stream>>>(
            h2, dinv, src, dst, agg2, N_EDGES);

        bias_relu_kernel<LAT><<<(int)((tot + T - 1) / T), T, 0, stream>>>(agg2, b2, out, N_NODES);
    }
}